// YOLOLoss_75247827026439
// MI455X (gfx1250) — compile-verified
//
#include <hip/hip_runtime.h>
#include <hip/hip_bf16.h>
#include <math.h>

typedef float v2f __attribute__((ext_vector_type(2)));
typedef float v8f __attribute__((ext_vector_type(8)));

#define NBATCH 16
#define NOBJ   30
#define NA     3
#define NC     80
#define NBLK   256   // blocks per scale in stageB (fixed -> deterministic partials)
#define TPB    256   // 8 waves of 32

// cells per scale: 16*3*76*76, 16*3*38*38, 16*3*19*19
#define S0 277248
#define S1 69312
#define S2 17328
#define SW (S0 + S1 + S2)          // 363888 total cells
#define WINNER_BYTES (SW * 4)      // int winner per cell
#define CLEAR_OFF  WINNER_BYTES    // 1455552
#define PART_OFF   (WINNER_BYTES + SW)          // 1819440 (SW % 4 == 0)
#define PART_FLOATS (3 * NBLK * 16)             // 12288
#define CNT_OFF    (PART_OFF + PART_FLOATS * 4) // 1868592
#define WS_WORDS   ((CNT_OFF + 8) / 4)          // words to zero

__device__ __constant__ float d_anch[9][2] = {
    {10.f,13.f},{16.f,30.f},{33.f,23.f},{30.f,61.f},{62.f,45.f},
    {59.f,119.f},{116.f,90.f},{156.f,198.f},{373.f,326.f}};

__device__ __forceinline__ float sigm(float x) { return 1.0f / (1.0f + expf(-x)); }

// ---------------- init: zero all workspace words ----------------
__global__ void yolo_init(int* ws, int nwords) {
    for (int t = blockIdx.x * blockDim.x + threadIdx.x; t < nwords;
         t += gridDim.x * blockDim.x)
        ws[t] = 0;
}

// ---------------- stage A: per-object scatter + stats ----------------
__global__ void yolo_stageA(const float* __restrict__ xs, const float* __restrict__ xm,
                            const float* __restrict__ xl, const float* __restrict__ boxes,
                            const int* __restrict__ cls, int* __restrict__ winner,
                            unsigned char* __restrict__ clr, int* __restrict__ counters) {
    const int s = blockIdx.x;          // scale
    const int t = threadIdx.x;
    if (t >= NBATCH * NOBJ) return;
    const int b = t / NOBJ, o = t % NOBJ;

    const int   ngA[3]  = {76, 38, 19};
    const float strA[3] = {8.f, 16.f, 32.f};
    const int   wOffA[3] = {0, S0, S0 + S1};
    const float* outp = (s == 0) ? xs : (s == 1) ? xm : xl;
    const int ng = ngA[s];
    const float stride = strA[s];
    const int wOff = wOffA[s];

    const float* bx = boxes + (b * NOBJ + o) * 4;
    const float g0 = bx[0] * ng, g1 = bx[1] * ng, g2 = bx[2] * ng, g3 = bx[3] * ng;
    const float w = g2, h = g3;

    float iou[9];
    int best = 0; float bestv = -1.0f;
    #pragma unroll
    for (int n = 0; n < 9; ++n) {
        float aw = d_anch[n][0] / stride, ah = d_anch[n][1] / stride;
        float inter = fminf(w, aw) * fminf(h, ah);
        iou[n] = inter / (w * h + aw * ah - inter + 1e-16f);
        if (iou[n] > bestv) { bestv = iou[n]; best = n; }   // first-max wins (jnp.argmax)
    }
    const bool bm = (best / 3) == s;
    const int a  = best % 3;
    const int gi = (int)g0, gj = (int)g1;

    if (bm) {
        atomicMax(&winner[wOff + ((b * NA + a) * ng + gj) * ng + gi], o + 1);
        #pragma unroll
        for (int k = 0; k < 3; ++k)
            if (iou[s * 3 + k] > 0.5f)                      // IGNORE_THRES
                clr[wOff + ((b * NA + k) * ng + gi) * ng + gj] = 1;   // note gi,gj transposed (ref quirk)
        atomicAdd(&counters[0], 1);                         // n_matched
    }

    // pred-box gather at (b, a, gj, gi)
    const long st = (long)ng * ng;
    const float* base = outp + (((long)b * 255 + (long)a * 85) * ng + gj) * ng + gi;
    const float o0 = base[0], o1 = base[st], o2 = base[2 * st], o3 = base[3 * st], o4 = base[4 * st];
    const float px = sigm(o0) + (float)gi, py = sigm(o1) + (float)gj;
    const float pw = expf(o2) * (d_anch[s * 3 + a][0] / stride);
    const float ph = expf(o3) * (d_anch[s * 3 + a][1] / stride);
    const float pc = sigm(o4);
    int pcls = 0; float pbv = base[5 * st];
    for (int c = 1; c < NC; ++c) {
        float v = base[(5 + c) * st];
        if (v > pbv) { pbv = v; pcls = c; }
    }
    const float gx1 = g0 - g2 * 0.5f, gy1 = g1 - g3 * 0.5f;
    const float gx2 = g0 + g2 * 0.5f, gy2 = g1 + g3 * 0.5f;
    // reference treats (x,y,w,h) as corners here — replicate
    const float wi = fmaxf(fminf(pw, gx2) - fmaxf(px, gx1), 0.f);
    const float hi = fmaxf(fminf(ph, gy2) - fmaxf(py, gy1), 0.f);
    const float inter2 = wi * hi;
    const float areap = (pw - px) * (ph - py);
    const float areag = (gx2 - gx1) * (gy2 - gy1);
    const float ioupg = inter2 / (areap + areag - inter2);
    if (bm && (ioupg > 0.5f) && (pc > 0.5f) && (pcls == cls[b * NOBJ + o]))
        atomicAdd(&counters[1], 1);                         // n_correct
}

// ---------------- stage B: per-cell pass (channel-4 plane + winner-cell gathers) ----------------
__global__ void __launch_bounds__(TPB)
yolo_stageB(const float* __restrict__ xs, const float* __restrict__ xm,
            const float* __restrict__ xl, const float* __restrict__ boxes,
            const int* __restrict__ cls, const int* __restrict__ winner,
            const unsigned char* __restrict__ clr, float* __restrict__ partials) {
    const int s = blockIdx.y;
    const int   ngA[3]  = {76, 38, 19};
    const float strA[3] = {8.f, 16.f, 32.f};
    const int   wOffA[3] = {0, S0, S0 + S1};
    const float* outp = (s == 0) ? xs : (s == 1) ? xm : xl;
    const int ng = ngA[s];
    const float stride = strA[s];
    const int wOff = wOffA[s];
    const int ncell = NBATCH * NA * ng * ng;

    // acc: 0 sxy, 1 swh, 2 sobj, 3 snoobj, 4 scls, 5 n_obj, 6 n_noobj, 7 n_prop
    float acc[8];
    #pragma unroll
    for (int q = 0; q < 8; ++q) acc[q] = 0.f;

    for (int t = blockIdx.x * blockDim.x + threadIdx.x; t < ncell;
         t += gridDim.x * blockDim.x) {
        const int i = t % ng;
        const int j = (t / ng) % ng;
        const int a = (t / (ng * ng)) % NA;
        const int b = t / (NA * ng * ng);
        const long st = (long)ng * ng;
        const float* base = outp + (((long)b * 255 + (long)a * 85) * ng + j) * ng + i;

        const float s4 = sigm(base[4 * st]);
        if (s4 > 0.5f) acc[7] += 1.0f;

        const int wv = winner[wOff + t];
        if (wv > 0) {                                    // object cell
            acc[5] += 1.0f;
            const float s0 = sigm(base[0]);
            const float s1 = sigm(base[st]);
            const float r2 = base[2 * st], r3 = base[3 * st];
            const int o = wv - 1;
            const float* bxp = boxes + (b * NOBJ + o) * 4;
            const float tw = logf(bxp[2] * ng / (d_anch[s * 3 + a][0] / stride) + 1e-16f);
            const float th = logf(bxp[3] * ng / (d_anch[s * 3 + a][1] / stride) + 1e-16f);
            const float dx = s0 - (float)i, dy = s1 - (float)j;   // target_xy = floor(gt) = (i, j)
            acc[0] += dx * dx + dy * dy;
            const float dw = r2 - tw, dh = r3 - th;
            acc[1] += dw * dw + dh * dh;
            acc[2] += -fmaxf(logf(s4), -100.0f);                  // bce(p, 1)
            const int cl = cls[b * NOBJ + o];
            float cs = 0.f;
            for (int c = 0; c < NC; ++c) {
                float sc = sigm(base[(5 + c) * st]);
                float lp = (c == cl) ? fmaxf(logf(sc), -100.f)
                                     : fmaxf(logf(1.0f - sc), -100.f);
                cs -= lp;
            }
            acc[4] += cs;
        } else if (clr[wOff + t] == 0) {                 // no-object cell
            acc[6] += 1.0f;
            acc[3] += -fmaxf(logf(1.0f - s4), -100.0f);  // bce(p, 0)
        }
    }

    // wave32 shuffle reduce, then LDS across 8 waves
    __shared__ float red[8][8];
    const int lane = threadIdx.x & 31;
    const int wid  = threadIdx.x >> 5;
    #pragma unroll
    for (int q = 0; q < 8; ++q) {
        float v = acc[q];
        #pragma unroll
        for (int m = 16; m >= 1; m >>= 1) v += __shfl_xor(v, m, 32);
        if (lane == 0) red[wid][q] = v;
    }
    __syncthreads();
    if (threadIdx.x < 8) {
        float sum = 0.f;
        #pragma unroll
        for (int w8 = 0; w8 < 8; ++w8) sum += red[w8][threadIdx.x];
        partials[((s * NBLK) + blockIdx.x) * 16 + threadIdx.x] = sum;
        // rows 8..15 of the quantity matrix stay zero (init kernel)
    }
}

// ---------------- finalize: WMMA f32 16x16x4 reduction over block partials ----------------
__global__ void yolo_finalize(const float* __restrict__ partials,
                              const int* __restrict__ counters,
                              float* __restrict__ out) {
    const int lane = threadIdx.x;        // exactly one wave32, EXEC all ones
    const int m  = lane & 15;            // quantity row
    const int k0 = (lane >> 4) * 2;      // A-layout: lanes<16 hold K=0,1; lanes>=16 hold K=2,3
    v2f bones; bones.x = 1.0f; bones.y = 1.0f;   // all-ones B: layout-independent

    float sums[3][8];
    for (int s = 0; s < 3; ++s) {
        v8f c = {0.f, 0.f, 0.f, 0.f, 0.f, 0.f, 0.f, 0.f};
        const float* base = partials + s * NBLK * 16;
        for (int chunk = 0; chunk < NBLK / 4; ++chunk) {
            v2f a;
            a.x = base[(chunk * 4 + k0 + 0) * 16 + m];
            a.y = base[(chunk * 4 + k0 + 1) * 16 + m];
            // D[m][n] += sum_k A[m][k] -> row sums of the 16x(4) partial tile
            c = __builtin_amdgcn_wmma_f32_16x16x4_f32(false, a, false, bones,
                                                      (short)0, c, false, false);
        }
        #pragma unroll
        for (int r = 0; r < 8; ++r) sums[s][r] = c[r];   // lane 0: D[r][0] = total of quantity r
    }

    if (lane == 0) {
        float lxy = 0.f, lwh = 0.f, lconf = 0.f, lcls = 0.f, nprop = 0.f;
        for (int s = 0; s < 3; ++s) {
            const float sxy = sums[s][0], swh = sums[s][1], sobj = sums[s][2];
            const float sno = sums[s][3], scl = sums[s][4];
            const float nob = sums[s][5], nno = sums[s][6];
            const float c2  = nob * 2.0f, c80 = nob * 80.0f;
            lxy  += (c2  > 0.f) ? sxy / fmaxf(c2, 1.f)  : 0.f;
            lwh  += (c2  > 0.f) ? swh / fmaxf(c2, 1.f)  : 0.f;
            lconf += 10.0f * ((nob > 0.f) ? sobj / fmaxf(nob, 1.f) : 0.f)
                   +  1.0f * ((nno > 0.f) ? sno  / fmaxf(nno, 1.f) : 0.f);
            lcls += (c80 > 0.f) ? scl / fmaxf(c80, 1.f) : 0.f;
            nprop += sums[s][7];
        }
        out[0] = lxy + lwh + lconf + lcls;
        out[1] = lxy;
        out[2] = lwh;
        out[3] = lconf;
        out[4] = lcls;
        out[5] = (float)counters[0];   // n_labels
        out[6] = nprop;                // n_proposals
        out[7] = (float)counters[1];   // n_correct
    }
}

extern "C" void kernel_launch(void* const* d_in, const int* in_sizes, int n_in,
                              void* d_out, int out_size, void* d_ws, size_t ws_size,
                              hipStream_t stream) {
    const float* xs    = (const float*)d_in[0];
    const float* xm    = (const float*)d_in[1];
    const float* xl    = (const float*)d_in[2];
    const float* boxes = (const float*)d_in[3];
    const int*   cls   = (const int*)d_in[4];
    float* out = (float*)d_out;

    char* wsb = (char*)d_ws;
    int*           winner   = (int*)wsb;
    unsigned char* clr      = (unsigned char*)(wsb + CLEAR_OFF);
    float*         partials = (float*)(wsb + PART_OFF);
    int*           counters = (int*)(wsb + CNT_OFF);

    yolo_init<<<256, 256, 0, stream>>>((int*)d_ws, WS_WORDS);
    yolo_stageA<<<3, 512, 0, stream>>>(xs, xm, xl, boxes, cls, winner, clr, counters);
    yolo_stageB<<<dim3(NBLK, 3), TPB, 0, stream>>>(xs, xm, xl, boxes, cls, winner, clr, partials);
    yolo_finalize<<<1, 32, 0, stream>>>(partials, counters, out);
}